// GNN_52252572123297
// MI455X (gfx1250) — compile-verified
//
#include <hip/hip_runtime.h>
#include <hip/hip_bf16.h>

// ---------------------------------------------------------------------------
// Problem constants (from the reference)
// ---------------------------------------------------------------------------
#define Bn        64
#define Nn        2048
#define En        32768
#define DIMn      64
#define K1n       1639
#define K2n       1312
#define NCLS      6
#define NTOT      (Bn * Nn)        // 131072 nodes
#define ETOT      (Bn * En)        // 2097152 edges
#define N1        (Bn * K1n)       // 104896
#define N2        (Bn * K2n)       // 83968
#define SORT_N    2048

typedef __attribute__((ext_vector_type(2))) float v2f;
typedef __attribute__((ext_vector_type(8))) float v8f;

// ---------------------------------------------------------------------------
// WMMA GEMM:  Y[nrows,64] = X[nrows,64] @ W[64,64]   (f32, exact)
// Each wave computes one 16x16 output tile; block = 256 threads = 8 waves
// = 2 row-tiles x 4 col-tiles = 32 rows / block.  nrows % 32 == 0.
// No divergence anywhere -> EXEC all ones (WMMA requirement).
// ---------------------------------------------------------------------------
__global__ __launch_bounds__(256)
void gemm64_wmma(const float* __restrict__ X, const float* __restrict__ W,
                 float* __restrict__ Y, int nrows) {
  const int lane = threadIdx.x & 31;
  const int wid  = threadIdx.x >> 5;                 // 0..7 (wave32)
  const int row0 = (blockIdx.x * 2 + (wid >> 2)) * 16;
  const int col0 = (wid & 3) * 16;
  const int m    = lane & 15;                        // A: row in tile; B/D: col
  const int hi   = lane >> 4;                        // lane half select

  v8f acc = {};
  const float* xrow = X + (size_t)(row0 + m) * DIMn;
  #pragma unroll
  for (int k0 = 0; k0 < 64; k0 += 4) {
    // A 16x4 f32: VGPR0 = K{0 | 2}, VGPR1 = K{1 | 3} per lane half
    const int ka = k0 + 2 * hi;
    v2f a, b;
    a.x = xrow[ka];
    a.y = xrow[ka + 1];
    // B 4x16 f32: rows of W striped across lanes, K split across lane halves
    b.x = W[(size_t)(ka    ) * DIMn + col0 + m];
    b.y = W[(size_t)(ka + 1) * DIMn + col0 + m];
    acc = __builtin_amdgcn_wmma_f32_16x16x4_f32(
        /*neg_a=*/false, a, /*neg_b=*/false, b,
        /*c_mod=*/(short)0, acc, /*reuse_a=*/false, /*reuse_b=*/false);
  }
  // C/D 16x16 f32 layout: VGPR r -> row (r + 8*hi), lane%16 -> col
  float* yb = Y + (size_t)row0 * DIMn + col0;
  #pragma unroll
  for (int r = 0; r < 8; ++r)
    yb[(size_t)(r + 8 * hi) * DIMn + m] = acc[r];
}

// ---------------------------------------------------------------------------
// Fill helpers
// ---------------------------------------------------------------------------
__global__ void fill_f32(float* __restrict__ p, float v, int n) {
  int i = blockIdx.x * blockDim.x + threadIdx.x;
  if (i < n) p[i] = v;
}

// ---------------------------------------------------------------------------
// Degree accumulate: deg[dst] += 1 for valid edges (invalid marked with -1)
// ---------------------------------------------------------------------------
__global__ void deg_accum(const int* __restrict__ src, const int* __restrict__ dst,
                          float* __restrict__ deg, int nedges) {
  int e = blockIdx.x * blockDim.x + threadIdx.x;
  if (e >= nedges) return;
  int s = src[e], d = dst[e];
  if (s >= 0 && d >= 0) atomicAdd(&deg[d], 1.0f);
}

// ---------------------------------------------------------------------------
// Edge scatter: agg[dst] += xw[src] * rsqrt(deg[src]*deg[dst])
// One wave per edge; each lane handles 2 channels (float2).
// Both xw and agg are ~32MB -> L2 resident (192MB L2).
// ---------------------------------------------------------------------------
__global__ __launch_bounds__(256)
void edge_scatter(const int* __restrict__ src, const int* __restrict__ dst,
                  const float* __restrict__ xw, const float* __restrict__ deg,
                  float* __restrict__ agg, int nedges) {
  long long idx = (long long)blockIdx.x * blockDim.x + threadIdx.x;
  int e = (int)(idx >> 5);
  if (e >= nedges) return;
  int c = ((int)idx & 31) * 2;
  int s = src[e], d = dst[e];
  if (s < 0 || d < 0) return;
  float nrm = rsqrtf(deg[s] * deg[d]);
  float2 v = *(const float2*)(xw + (size_t)s * DIMn + c);
  atomicAdd(agg + (size_t)d * DIMn + c,     v.x * nrm);
  atomicAdd(agg + (size_t)d * DIMn + c + 1, v.y * nrm);
}

// ---------------------------------------------------------------------------
// Finalize (in place): agg[i] = relu(agg[i] + xw[i]/deg[row] + bias[ch])
// ---------------------------------------------------------------------------
__global__ void gcn_finalize(float* __restrict__ agg, const float* __restrict__ xw,
                             const float* __restrict__ deg, const float* __restrict__ bias,
                             int ntot) {
  int i = blockIdx.x * blockDim.x + threadIdx.x;
  if (i >= ntot) return;
  int c = i & (DIMn - 1);
  int r = i >> 6;
  float v = agg[i] + xw[i] / deg[r] + bias[c];
  agg[i] = v > 0.0f ? v : 0.0f;
}

// ---------------------------------------------------------------------------
// TopK pooling: per-graph scores = tanh(h.p/||p||); bitonic sort 2048
// (score,idx) pairs in LDS (descending); emit gated rows xp = h[sel]*score,
// optional new_id remap (pool1).  grid = B, block = 1024 (32 waves).
// ---------------------------------------------------------------------------
__global__ __launch_bounds__(1024)
void topk_pool(const float* __restrict__ h, const float* __restrict__ p,
               float* __restrict__ xp, int* __restrict__ new_id,
               int n_per, int k) {
  __shared__ float ss[SORT_N];
  __shared__ int   si[SORT_N];
  const int g = blockIdx.x;
  const int t = threadIdx.x;

  float nrm = 0.0f;
  #pragma unroll
  for (int i = 0; i < DIMn; ++i) { float pv = p[i]; nrm += pv * pv; }
  const float rinv = rsqrtf(nrm);

  const float* hg = h + (size_t)g * n_per * DIMn;
  for (int i = t; i < SORT_N; i += 1024) {
    float sc;
    if (i < n_per) {
      const float* row = hg + (size_t)i * DIMn;
      float dot = 0.0f;
      #pragma unroll
      for (int c = 0; c < DIMn; ++c) dot += row[c] * p[c];
      sc = tanhf(dot * rinv);
      if (new_id) new_id[(size_t)g * n_per + i] = -1;
    } else {
      sc = -INFINITY;  // padding sorts to the bottom
    }
    ss[i] = sc;
    si[i] = i;
  }
  __syncthreads();

  // Bitonic sort, descending. Each XOR-pair owned by exactly one thread-iter.
  for (int kk = 2; kk <= SORT_N; kk <<= 1) {
    for (int j = kk >> 1; j > 0; j >>= 1) {
      for (int i = t; i < SORT_N; i += 1024) {
        int ixj = i ^ j;
        if (ixj > i) {
          bool desc = ((i & kk) == 0);
          float a = ss[i], b = ss[ixj];
          if ((a < b) == desc) {
            ss[i] = b; ss[ixj] = a;
            int tmp = si[i]; si[i] = si[ixj]; si[ixj] = tmp;
          }
        }
      }
      __syncthreads();
    }
  }

  // Emit top-k gated rows (+ remap). Per-thread contiguous 256B row copies.
  for (int j = t; j < k; j += 1024) {
    float val = ss[j];
    int loc = si[j];
    const float* row = hg + (size_t)loc * DIMn;
    float* orow = xp + ((size_t)g * k + j) * DIMn;
    #pragma unroll 16
    for (int c = 0; c < DIMn; ++c) orow[c] = row[c] * val;
    if (new_id) new_id[(size_t)g * n_per + loc] = g * k + j;
  }
}

// ---------------------------------------------------------------------------
// Per-graph channelwise [max, mean] over k rows.  grid = B, block = 1024.
// ---------------------------------------------------------------------------
__global__ __launch_bounds__(1024)
void reduce_max_mean(const float* __restrict__ xp, float* __restrict__ out, int k) {
  __shared__ float smax[16][DIMn];
  __shared__ float ssum[16][DIMn];
  const int g = blockIdx.x;
  const int c = threadIdx.x & 63;
  const int r = threadIdx.x >> 6;   // 0..15
  float mx = -INFINITY, sm = 0.0f;
  for (int j = r; j < k; j += 16) {
    float v = xp[((size_t)g * k + j) * DIMn + c];
    mx = fmaxf(mx, v);
    sm += v;
  }
  smax[r][c] = mx;
  ssum[r][c] = sm;
  __syncthreads();
  if (threadIdx.x < 64) {
    float m2 = smax[0][c], s2 = ssum[0][c];
    #pragma unroll
    for (int rr = 1; rr < 16; ++rr) { m2 = fmaxf(m2, smax[rr][c]); s2 += ssum[rr][c]; }
    out[(size_t)g * (2 * DIMn) + c]        = m2;
    out[(size_t)g * (2 * DIMn) + DIMn + c] = s2 / (float)k;
  }
}

// ---------------------------------------------------------------------------
// Remap edges to pooled node ids; edges touching dropped nodes -> (-1,-1)
// ---------------------------------------------------------------------------
__global__ void remap_edges(const int* __restrict__ src, const int* __restrict__ dst,
                            const int* __restrict__ new_id,
                            int* __restrict__ s2, int* __restrict__ d2, int nedges) {
  int e = blockIdx.x * blockDim.x + threadIdx.x;
  if (e >= nedges) return;
  int a = new_id[src[e]];
  int b = new_id[dst[e]];
  if (a < 0 || b < 0) { a = -1; b = -1; }
  s2[e] = a;
  d2[e] = b;
}

// ---------------------------------------------------------------------------
// MLP head: h = relu((x1+x2) @ lin1_w + lin1_b); out = h @ lin2_w + lin2_b
// ---------------------------------------------------------------------------
__global__ __launch_bounds__(64)
void head1(const float* __restrict__ x1, const float* __restrict__ x2,
           const float* __restrict__ w, const float* __restrict__ b,
           float* __restrict__ hmid) {
  __shared__ float hin[2 * DIMn];
  const int g = blockIdx.x, t = threadIdx.x;
  hin[t]        = x1[g * 128 + t]      + x2[g * 128 + t];
  hin[t + 64]   = x1[g * 128 + 64 + t] + x2[g * 128 + 64 + t];
  __syncthreads();
  float acc = b[t];
  #pragma unroll 8
  for (int k = 0; k < 128; ++k) acc += hin[k] * w[k * DIMn + t];
  hmid[g * DIMn + t] = acc > 0.0f ? acc : 0.0f;
}

__global__ void head2(const float* __restrict__ hmid, const float* __restrict__ w,
                      const float* __restrict__ b, float* __restrict__ out) {
  int idx = blockIdx.x * blockDim.x + threadIdx.x;
  if (idx >= Bn * NCLS) return;
  int g = idx / NCLS, c = idx % NCLS;
  float acc = b[c];
  #pragma unroll 8
  for (int j = 0; j < DIMn; ++j) acc += hmid[g * DIMn + j] * w[j * NCLS + c];
  out[idx] = acc;
}

// ---------------------------------------------------------------------------
// Orchestration
// ---------------------------------------------------------------------------
extern "C" void kernel_launch(void* const* d_in, const int* in_sizes, int n_in,
                              void* d_out, int out_size, void* d_ws, size_t ws_size,
                              hipStream_t stream) {
  const float* x      = (const float*)d_in[0];
  const int*   eidx   = (const int*)  d_in[1];   // [2, ETOT]
  const float* W1     = (const float*)d_in[3];
  const float* b1     = (const float*)d_in[4];
  const float* p1     = (const float*)d_in[5];
  const float* W2     = (const float*)d_in[6];
  const float* b2     = (const float*)d_in[7];
  const float* p2     = (const float*)d_in[8];
  const float* lin1w  = (const float*)d_in[9];
  const float* lin1b  = (const float*)d_in[10];
  const float* lin2w  = (const float*)d_in[11];
  const float* lin2b  = (const float*)d_in[12];
  float* out = (float*)d_out;

  const int* src = eidx;
  const int* dst = eidx + ETOT;

  // Workspace carve-up (buffers recycled between layer 1 and layer 2)
  char* ws = (char*)d_ws;
  size_t off = 0;
  auto carve = [&](size_t bytes) { void* p = ws + off; off += (bytes + 1023) & ~(size_t)1023; return p; };
  float* buf_xw  = (float*)carve((size_t)NTOT * DIMn * sizeof(float)); // xw1, then xw2
  float* buf_agg = (float*)carve((size_t)NTOT * DIMn * sizeof(float)); // agg1->h1, agg2->h2
  float* buf_xp  = (float*)carve((size_t)N1   * DIMn * sizeof(float)); // xp1, then xp2
  float* deg     = (float*)carve((size_t)NTOT * sizeof(float));        // deg1, then deg2
  int*   new_id  = (int*)  carve((size_t)NTOT * sizeof(int));
  int*   s2      = (int*)  carve((size_t)ETOT * sizeof(int));
  int*   d2      = (int*)  carve((size_t)ETOT * sizeof(int));
  float* x1v     = (float*)carve((size_t)Bn * 128 * sizeof(float));
  float* x2v     = (float*)carve((size_t)Bn * 128 * sizeof(float));
  float* hmid    = (float*)carve((size_t)Bn * DIMn * sizeof(float));
  (void)ws_size; (void)in_sizes; (void)n_in; (void)out_size;

  const int T = 256;

  // ---------------- Layer 1 ----------------
  fill_f32<<<(NTOT + T - 1) / T, T, 0, stream>>>(deg, 1.0f, NTOT);
  deg_accum<<<(ETOT + T - 1) / T, T, 0, stream>>>(src, dst, deg, ETOT);

  gemm64_wmma<<<NTOT / 32, 256, 0, stream>>>(x, W1, buf_xw, NTOT);          // xw1

  fill_f32<<<(NTOT * DIMn + T - 1) / T, T, 0, stream>>>(buf_agg, 0.0f, NTOT * DIMn);
  edge_scatter<<<(int)(((long long)ETOT * 32 + T - 1) / T), T, 0, stream>>>(
      src, dst, buf_xw, deg, buf_agg, ETOT);
  gcn_finalize<<<(NTOT * DIMn + T - 1) / T, T, 0, stream>>>(buf_agg, buf_xw, deg, b1,
                                                            NTOT * DIMn);  // h1 in buf_agg

  topk_pool<<<Bn, 1024, 0, stream>>>(buf_agg, p1, buf_xp, new_id, Nn, K1n); // xp1
  reduce_max_mean<<<Bn, 1024, 0, stream>>>(buf_xp, x1v, K1n);
  remap_edges<<<(ETOT + T - 1) / T, T, 0, stream>>>(src, dst, new_id, s2, d2, ETOT);

  // ---------------- Layer 2 ----------------
  gemm64_wmma<<<N1 / 32, 256, 0, stream>>>(buf_xp, W2, buf_xw, N1);         // xw2

  fill_f32<<<(N1 + T - 1) / T, T, 0, stream>>>(deg, 1.0f, N1);
  deg_accum<<<(ETOT + T - 1) / T, T, 0, stream>>>(s2, d2, deg, ETOT);

  fill_f32<<<(N1 * DIMn + T - 1) / T, T, 0, stream>>>(buf_agg, 0.0f, N1 * DIMn);
  edge_scatter<<<(int)(((long long)ETOT * 32 + T - 1) / T), T, 0, stream>>>(
      s2, d2, buf_xw, deg, buf_agg, ETOT);
  gcn_finalize<<<(N1 * DIMn + T - 1) / T, T, 0, stream>>>(buf_agg, buf_xw, deg, b2,
                                                          N1 * DIMn);       // h2 in buf_agg

  topk_pool<<<Bn, 1024, 0, stream>>>(buf_agg, p2, buf_xp, (int*)nullptr, K1n, K2n); // xp2
  reduce_max_mean<<<Bn, 1024, 0, stream>>>(buf_xp, x2v, K2n);

  // ---------------- Head ----------------
  head1<<<Bn, 64, 0, stream>>>(x1v, x2v, lin1w, lin1b, hmid);
  head2<<<(Bn * NCLS + 63) / 64, 64, 0, stream>>>(hmid, lin2w, lin2b, out);
}